// MyModel_61933428411925
// MI455X (gfx1250) — compile-verified
//
#include <hip/hip_runtime.h>

// CDNA5 / gfx1250: wave32, WMMA 16x16x4 F32 (fp32-exact matrix pipe).
typedef float v2f __attribute__((ext_vector_type(2)));
typedef float v8f __attribute__((ext_vector_type(8)));

#define DIM_D 256            // row length (reduction dim)
#define DIM_N 1024           // rows per batch
#define EDGES_PER_WAVE 16
#define WAVES_PER_BLOCK 8
#define EDGES_PER_BLOCK (EDGES_PER_WAVE * WAVES_PER_BLOCK)

__global__ __launch_bounds__(256, 4)
void edge_norm_wmma(const float* __restrict__ x,
                    const int*   __restrict__ eb,
                    const int*   __restrict__ ei,
                    const int*   __restrict__ ej,
                    float*       __restrict__ out,
                    int E)
{
    // 16x16 fp32 C tile per wave for diagonal extraction (1 KB/wave)
    __shared__ float lds[WAVES_PER_BLOCK][16 * 16];

    const int tid   = threadIdx.x;
    const int wave  = tid >> 5;
    const int lane  = tid & 31;
    const int m     = lane & 15;         // edge slot within the wave's 16-edge group
    const int khalf = (lane >> 4) * 2;   // K offset inside each 4-wide chunk (A layout)

    const long groupBase = (long)blockIdx.x * EDGES_PER_BLOCK
                         + (long)wave * EDGES_PER_WAVE;
    const long e  = groupBase + m;
    const long ec = (e < (long)E) ? e : (long)(E - 1);  // clamp loads; EXEC stays all-1s

    const int b = eb[ec];
    const int i = ei[ec];
    const int j = ej[ec];
    const float* __restrict__ pi = x + ((size_t)b * DIM_N + (size_t)i) * DIM_D;
    const float* __restrict__ pj = x + ((size_t)b * DIM_N + (size_t)j) * DIM_D;

    v8f c = {0.f, 0.f, 0.f, 0.f, 0.f, 0.f, 0.f, 0.f};

    // 64 accumulating WMMAs: D = A * A^T + C.  A = 16 edges x 4 K of diff.
    // f32 A layout: lane l -> (M = l&15, K = (l>>4)*2 + {0,1});
    // f32 B layout is the lane/N mirror, so passing the same regs as B yields
    // the Gram matrix of the diff vectors; its diagonal is ||diff_m||^2.
    #pragma unroll 4
    for (int kb = 0; kb < DIM_D; kb += 4) {
        const int k = kb + khalf;                    // 8-byte aligned
        v2f xi = *(const v2f*)(pi + k);              // global_load_b64
        v2f xj = *(const v2f*)(pj + k);              // global_load_b64
        v2f a;
        a.x = xi.x - xj.x;
        a.y = xi.y - xj.y;
        c = __builtin_amdgcn_wmma_f32_16x16x4_f32(
                /*neg_a=*/false, a, /*neg_b=*/false, a,
                /*c_mod=*/(short)0, c, /*reuse_a=*/false, /*reuse_b=*/false);
    }

    // C/D layout: VGPR r, lanes 0-15 -> (M=r,   N=lane);
    //             lanes 16-31        -> (M=r+8, N=lane-16).
    float* tile = lds[wave];
    const int nIdx  = lane & 15;
    const int mBase = (lane >> 4) * 8;
    #pragma unroll
    for (int r = 0; r < 8; ++r)
        tile[(mBase + r) * 16 + nIdx] = c[r];

    __syncthreads();   // orders the cross-lane LDS bounce (uniform, EXEC all-1s)

    if (lane < 16 && e < (long)E) {
        const float s = tile[m * 16 + m];            // diagonal: banks 17m%64, no conflict
        const float d = sqrtf(s);                    // abs(sqrt(.)) == sqrt(.) for s>=0
        float4 v = make_float4(d, d, d, d);
        float4* o = (float4*)(out + (size_t)e * 8);  // 8-fold broadcast, two b128 stores
        o[0] = v;
        o[1] = v;
    }
}

extern "C" void kernel_launch(void* const* d_in, const int* in_sizes, int n_in,
                              void* d_out, int out_size, void* d_ws, size_t ws_size,
                              hipStream_t stream)
{
    const float* x  = (const float*)d_in[0];
    const int*   eb = (const int*)d_in[1];
    const int*   ei = (const int*)d_in[2];
    const int*   ej = (const int*)d_in[3];
    float*       out = (float*)d_out;

    const int E = in_sizes[1];          // edge count (data-dependent)
    if (E <= 0) return;

    const int blocks = (E + EDGES_PER_BLOCK - 1) / EDGES_PER_BLOCK;
    hipLaunchKernelGGL(edge_norm_wmma, dim3(blocks), dim3(256), 0, stream,
                       x, eb, ei, ej, out, E);
}